// InteractionArch_36292473651202
// MI455X (gfx1250) — compile-verified
//
#include <hip/hip_runtime.h>

typedef float v2f __attribute__((ext_vector_type(2)));
typedef float v8f __attribute__((ext_vector_type(8)));

// Problem constants (from reference): B=16384, F=26, D=128
constexpr int kBatch  = 16384;
constexpr int kF      = 26;
constexpr int kD      = 128;
constexpr int kRows   = kF + 1;          // 27 rows in "combined"
constexpr int kTri    = kRows * kF / 2;  // 351 strict-upper-tri elements
constexpr int kOutW   = kD + kTri;       // 479 floats per batch
constexpr int kLStr   = 132;             // LDS row stride (floats): 132 % 64 == 4 -> conflict-free frag loads
constexpr int kGStr   = 28;              // Gram scratch row stride
constexpr int kWaves  = 2;               // waves (=batches) per workgroup

__global__ __launch_bounds__(32 * kWaves)
void dlrm_interact_wmma_f32(const float* __restrict__ dense,
                            const float* __restrict__ sparse,
                            float* __restrict__ out)
{
    __shared__ float comb[kWaves][32 * kLStr];       // padded 32x128 combined tile
    __shared__ float gram[kWaves][kRows * kGStr];    // 27x27 Gram scratch

    const int wave = threadIdx.x >> 5;
    const int lane = threadIdx.x & 31;
    const int b    = blockIdx.x * kWaves + wave;

    float* cb = comb[wave];
    float* gm = gram[wave];

    // ---- Stage combined (27x128 fp32) into LDS; fully coalesced float4 loads ----
    {
        float4 dv = *(const float4*)(dense + (size_t)b * kD + lane * 4);
        *(float4*)&cb[0 * kLStr + lane * 4] = dv;
        const float* srow = sparse + (size_t)b * kF * kD + lane * 4;
        #pragma unroll 4
        for (int i = 1; i < kRows; ++i) {
            float4 v = *(const float4*)(srow + (size_t)(i - 1) * kD);
            *(float4*)&cb[i * kLStr + lane * 4] = v;
        }
        float4 z = make_float4(0.f, 0.f, 0.f, 0.f);
        #pragma unroll
        for (int i = kRows; i < 32; ++i)
            *(float4*)&cb[i * kLStr + lane * 4] = z;
    }
    // Per-wave private LDS region; same-wave DS ops are in order -> no barrier needed.

    // ---- Gram = combined * combined^T via V_WMMA_F32_16X16X4_F32 ----
    // f32 A operand layout: lane(0..15)=row M, lanes 16..31 carry K+2; 2 VGPRs = 2 K values.
    // B operand is the mirror with lane = col N; since B = A^T here, the SAME registers
    // serve as both operands.
    const int l  = lane & 15;
    const int h2 = (lane >> 4) * 2;
    const float* p0 = &cb[(0  + l) * kLStr + h2];  // row block 0 (rows 0..15)
    const float* p1 = &cb[(16 + l) * kLStr + h2];  // row block 1 (rows 16..31, 27+ zeroed)

    v8f acc00 = {};  // G[0:16, 0:16]
    v8f acc01 = {};  // G[0:16, 16:32]
    v8f acc11 = {};  // G[16:32,16:32]

    #pragma unroll 8
    for (int kk = 0; kk < kD / 4; ++kk) {
        v2f f0 = *(const v2f*)(p0 + kk * 4);       // ds_load_b64, bank-conflict-free
        v2f f1 = *(const v2f*)(p1 + kk * 4);
        acc00 = __builtin_amdgcn_wmma_f32_16x16x4_f32(false, f0, false, f0, (short)0, acc00, false, false);
        acc01 = __builtin_amdgcn_wmma_f32_16x16x4_f32(false, f0, false, f1, (short)0, acc01, false, false);
        acc11 = __builtin_amdgcn_wmma_f32_16x16x4_f32(false, f1, false, f1, (short)0, acc11, false, false);
    }

    // ---- Spill the three needed tiles to LDS Gram scratch ----
    // C/D layout: lanes 0..15 -> N=lane, VGPR v -> M=v ; lanes 16..31 -> N=lane-16, M=v+8.
    {
        const int nl = lane & 15;
        const int mb = (lane >> 4) * 8;
        #pragma unroll
        for (int v = 0; v < 8; ++v)                     // tile (0,0)
            gm[(mb + v) * kGStr + nl] = acc00[v];
        if (16 + nl < kRows) {
            #pragma unroll
            for (int v = 0; v < 8; ++v)                 // tile (0,1)
                gm[(mb + v) * kGStr + 16 + nl] = acc01[v];
            #pragma unroll
            for (int v = 0; v < 8; ++v) {               // tile (1,1)
                int m = 16 + mb + v;
                if (m < kRows) gm[m * kGStr + 16 + nl] = acc11[v];
            }
        }
    }

    // ---- Write output: [dense (128) | strict upper triangle (351)] ----
    float* ob = out + (size_t)b * kOutW;
    #pragma unroll
    for (int t = lane; t < kD; t += 32)
        ob[t] = cb[t];                                  // combined row 0 == dense passthrough
    for (int t = lane; t < kTri; t += 32) {
        // invert row-major triu(k=1) index: t -> (i, j), i < j
        int i = 0, base = 0, cnt = kF;
        while (t - base >= cnt) { base += cnt; --cnt; ++i; }
        int j = i + 1 + (t - base);
        ob[kD + t] = gm[i * kGStr + j];
    }
}

extern "C" void kernel_launch(void* const* d_in, const int* in_sizes, int n_in,
                              void* d_out, int out_size, void* d_ws, size_t ws_size,
                              hipStream_t stream) {
    const float* dense  = (const float*)d_in[0];   // (16384, 128) fp32
    const float* sparse = (const float*)d_in[1];   // (16384, 26, 128) fp32
    float* out = (float*)d_out;                    // (16384, 479) fp32

    dim3 grid(kBatch / kWaves);
    dim3 block(32 * kWaves);
    hipLaunchKernelGGL(dlrm_interact_wmma_f32, grid, block, 0, stream,
                       dense, sparse, out);
}